// ResGCN_59880434041371
// MI455X (gfx1250) — compile-verified
//
#include <hip/hip_runtime.h>
#include <hip/hip_bf16.h>
#include <math.h>

#define DH 128
#define CLS 10
#define EPSV 1e-5f

typedef float v2f __attribute__((ext_vector_type(2)));
typedef float v8f __attribute__((ext_vector_type(8)));

// ---------------------------------------------------------------- utilities

__global__ void zero_kernel(float* __restrict__ p, long n4) {
    // n4 = number of float4 chunks (buffer sizes here are all multiples of 4)
    long i = (long)blockIdx.x * blockDim.x + threadIdx.x;
    if (i < n4) {
        ((float4*)p)[i] = make_float4(0.f, 0.f, 0.f, 0.f);
    }
}

__global__ void edge_deg_kernel(const int* __restrict__ src, const int* __restrict__ dst,
                                float* __restrict__ dsrc, float* __restrict__ ddst, int E) {
    int e = blockIdx.x * blockDim.x + threadIdx.x;
    if (e < E) {
        unsafeAtomicAdd(&dsrc[src[e]], 1.0f);
        unsafeAtomicAdd(&ddst[dst[e]], 1.0f);
    }
}

__global__ void deg_finalize_kernel(float* __restrict__ d, int n) {
    int i = blockIdx.x * blockDim.x + threadIdx.x;
    if (i < n) {
        float v = d[i];
        v = v < 1.0f ? 1.0f : v;
        d[i] = rsqrtf(v);
    }
}

// ------------------------------------------------------------ batch-norm ops

// 128 threads per block, one column per thread; block handles a row chunk.
__global__ void bn_stats_kernel(const float* __restrict__ x, int rows,
                                float* __restrict__ sum, float* __restrict__ sumsq) {
    int col = threadIdx.x;
    int chunk = (rows + gridDim.x - 1) / gridDim.x;
    int r0 = blockIdx.x * chunk;
    int r1 = r0 + chunk; if (r1 > rows) r1 = rows;
    float s = 0.f, s2 = 0.f;
    for (int r = r0; r < r1; ++r) {
        float v = x[(long)r * DH + col];
        s += v; s2 += v * v;
    }
    unsafeAtomicAdd(&sum[col], s);
    unsafeAtomicAdd(&sumsq[col], s2);
}

// y = x*colscale + colshift  ==  (x-mean)*rsqrt(var+eps)*g + b
__global__ void bn_finalize_kernel(const float* __restrict__ sum, const float* __restrict__ sumsq,
                                   const float* __restrict__ g, const float* __restrict__ b,
                                   int rows, float* __restrict__ colscale, float* __restrict__ colshift) {
    int c = threadIdx.x;
    float inv = 1.0f / (float)rows;
    float m   = sum[c] * inv;
    float var = sumsq[c] * inv - m * m;
    float sc  = g[c] * rsqrtf(var + EPSV);
    colscale[c] = sc;
    colshift[c] = b[c] - m * sc;
}

// ------------------------------------------------------- edge scatter (msg)
//
// one wave (32 lanes) per edge; lane handles 4 consecutive columns (float4).
// norm_dst[d] is folded into the per-edge factor, so the following GEMM needs
// no row scaling:  msg[d] += BN(h[s]) * (norm_src[s] * norm_dst[d])

__global__ __launch_bounds__(256) void scatter_kernel(
        const float* __restrict__ h, const int* __restrict__ src, const int* __restrict__ dst,
        const float* __restrict__ colscale, const float* __restrict__ colshift,
        const float* __restrict__ norm_src, const float* __restrict__ norm_dst,
        float* __restrict__ msg, int E) {
    int e = blockIdx.x * 8 + (threadIdx.x >> 5);
    if (e >= E) return;
    int lane = threadIdx.x & 31;
    int s = src[e];
    int d = dst[e];
    float f = norm_src[s] * norm_dst[d];
    const float4 hv = *(const float4*)(h + (long)s * DH + lane * 4);
    const float4 cs = *(const float4*)(colscale + lane * 4);
    const float4 cf = *(const float4*)(colshift + lane * 4);
    float* mp = msg + (long)d * DH + lane * 4;
    unsafeAtomicAdd(mp + 0, (hv.x * cs.x + cf.x) * f);
    unsafeAtomicAdd(mp + 1, (hv.y * cs.y + cf.y) * f);
    unsafeAtomicAdd(mp + 2, (hv.z * cs.z + cf.z) * f);
    unsafeAtomicAdd(mp + 3, (hv.w * cs.w + cf.w) * f);
}

// ------------------------------------------------------------- sum pooling

__global__ __launch_bounds__(256) void pool_kernel(const float* __restrict__ h,
                                                   const int* __restrict__ gid,
                                                   float* __restrict__ hg, int n) {
    int node = blockIdx.x * 8 + (threadIdx.x >> 5);
    if (node >= n) return;
    int lane = threadIdx.x & 31;
    int g = gid[node];
    const float4 v = *(const float4*)(h + (long)node * DH + lane * 4);
    float* p = hg + (long)g * DH + lane * 4;
    unsafeAtomicAdd(p + 0, v.x);
    unsafeAtomicAdd(p + 1, v.y);
    unsafeAtomicAdd(p + 2, v.z);
    unsafeAtomicAdd(p + 3, v.w);
}

// ---------------------------------------------------------------- WMMA GEMM
//
// out[rows,128] = relu( (A*colscale + colshift) @ W + bias )  [+ residual]
// rows must be a multiple of 16 (100000 and 512 both are).
// One 16x16 C tile per wave; 8 waves per block cover all 128 output columns
// of one 16-row strip. K=128 stepped by 4 with V_WMMA_F32_16X16X4_F32.
// AFFINE / RESIDUAL are compile-time so the hot loop carries no branches.

template <bool AFFINE, bool RESIDUAL>
__global__ __launch_bounds__(256) void gemm_wmma_kernel(
        const float* __restrict__ A, const float* __restrict__ W,
        const float* __restrict__ bias,
        const float* __restrict__ colscale, const float* __restrict__ colshift,
        const float* __restrict__ residual,
        float* __restrict__ out, int rows) {
    const int lane = threadIdx.x & 31;
    const int wave = threadIdx.x >> 5;      // column tile 0..7
    const int half = lane >> 4;             // lane group (K-split)
    const int l16  = lane & 15;
    const int tileRow = blockIdx.x * 16;
    const int arow = tileRow + l16;         // A row held by this lane (both halves)
    const int colB = wave * 16 + l16;       // B / C column held by this lane

    const float* __restrict__ ap = A + (long)arow * DH;

    v8f c = {0.f, 0.f, 0.f, 0.f, 0.f, 0.f, 0.f, 0.f};

#if __has_builtin(__builtin_amdgcn_wmma_f32_16x16x4_f32)
    #pragma unroll
    for (int k0 = 0; k0 < DH; k0 += 4) {
        const int ka = k0 + 2 * half;       // this lane's K pair: (ka, ka+1)
        v2f a = { ap[ka], ap[ka + 1] };     // contiguous -> global_load_b64
        if (AFFINE) {
            const float2 cs = *(const float2*)(colscale + ka);
            const float2 cf = *(const float2*)(colshift + ka);
            a[0] = a[0] * cs.x + cf.x;
            a[1] = a[1] * cs.y + cf.y;
        }
        v2f b = { W[(long)ka * DH + colB], W[(long)(ka + 1) * DH + colB] };
        c = __builtin_amdgcn_wmma_f32_16x16x4_f32(
                /*neg_a=*/false, a, /*neg_b=*/false, b,
                /*c_mod=*/(short)0, c, /*reuse_a=*/false, /*reuse_b=*/false);
    }
#else
    // scalar fallback matching the C fragment layout
    for (int i = 0; i < 8; ++i) {
        int r = tileRow + i + 8 * half;
        const float* ar = A + (long)r * DH;
        float acc = 0.f;
        for (int k = 0; k < DH; ++k) {
            float av = ar[k];
            if (AFFINE) av = av * colscale[k] + colshift[k];
            acc += av * W[(long)k * DH + colB];
        }
        c[i] = acc;
    }
#endif

    const float bv = bias[colB];
    #pragma unroll
    for (int i = 0; i < 8; ++i) {
        int r = tileRow + i + 8 * half;     // C layout: VGPR i -> row i (+8 for hi half)
        float v = c[i] + bv;
        v = v > 0.f ? v : 0.f;              // relu
        if (RESIDUAL) v += residual[(long)r * DH + colB];
        out[(long)r * DH + colB] = v;
    }
}

// --------------------------------------------------------- classifier head

__global__ void head_kernel(const float* __restrict__ hg2,
                            const float* __restrict__ colscale, const float* __restrict__ colshift,
                            const float* __restrict__ clsW, const float* __restrict__ clsb,
                            float* __restrict__ out, int G) {
    int g = blockIdx.x;
    int cth = threadIdx.x;
    if (g >= G || cth >= CLS) return;
    const float* hr = hg2 + (long)g * DH;
    float acc = clsb[cth];
    for (int k = 0; k < DH; ++k) {
        acc += (hr[k] * colscale[k] + colshift[k]) * clsW[k * CLS + cth];
    }
    out[(long)g * CLS + cth] = acc;
}

// ---------------------------------------------------------------- launcher

extern "C" void kernel_launch(void* const* d_in, const int* in_sizes, int n_in,
                              void* d_out, int out_size, void* d_ws, size_t ws_size,
                              hipStream_t stream) {
    const float* h_in        = (const float*)d_in[0];
    const float* bn_feat_g   = (const float*)d_in[1];
    const float* bn_feat_b   = (const float*)d_in[2];
    const float* feat_W      = (const float*)d_in[3];
    const float* feat_b      = (const float*)d_in[4];
    const float* conv_bn_g   = (const float*)d_in[5];
    const float* conv_bn_b   = (const float*)d_in[6];
    const float* conv_W      = (const float*)d_in[7];
    const float* conv_b      = (const float*)d_in[8];
    const float* fc_bn_g     = (const float*)d_in[9];
    const float* fc_bn_b     = (const float*)d_in[10];
    const float* fc_W        = (const float*)d_in[11];
    const float* fc_b        = (const float*)d_in[12];
    const float* bn_hidden_g = (const float*)d_in[13];
    const float* bn_hidden_b = (const float*)d_in[14];
    const float* cls_W       = (const float*)d_in[15];
    const float* cls_b       = (const float*)d_in[16];
    const int*   src         = (const int*)d_in[17];
    const int*   dst         = (const int*)d_in[18];
    const int*   gid         = (const int*)d_in[19];

    const int N = in_sizes[0] / DH;
    const int E = in_sizes[17];
    const int G = out_size / CLS;
    const int L = in_sizes[5] / DH;

    // ---- workspace carve-up (floats)
    float* ws       = (float*)d_ws;
    float* norm_src = ws;  ws += N;
    float* norm_dst = ws;  ws += N;
    float* sum      = ws;  ws += DH;
    float* sumsq    = ws;  ws += DH;
    float* colscale = ws;  ws += DH;
    float* colshift = ws;  ws += DH;
    float* hbuf     = ws;  ws += (size_t)N * DH;
    float* msg      = ws;  ws += (size_t)N * DH;
    float* hg       = ws;  ws += (size_t)G * DH;
    float* hg2     = ws;  ws += (size_t)G * DH;

    auto zero = [&](float* p, long nfloats) {
        long n4 = nfloats / 4;
        long blocks = (n4 + 255) / 256;
        zero_kernel<<<(unsigned)blocks, 256, 0, stream>>>(p, n4);
    };

    // ---- degree norms (norm_src / norm_dst are contiguous: zero both at once)
    zero(norm_src, 2L * N);
    edge_deg_kernel<<<(E + 255) / 256, 256, 0, stream>>>(src, dst, norm_src, norm_dst, E);
    deg_finalize_kernel<<<(2 * N + 255) / 256, 256, 0, stream>>>(norm_src, 2 * N);

    // ---- feature layer: h = relu(BN(h_in) @ feat_W + feat_b)
    zero(sum, 2L * DH);
    bn_stats_kernel<<<256, DH, 0, stream>>>(h_in, N, sum, sumsq);
    bn_finalize_kernel<<<1, DH, 0, stream>>>(sum, sumsq, bn_feat_g, bn_feat_b, N, colscale, colshift);
    gemm_wmma_kernel<true, false><<<N / 16, 256, 0, stream>>>(
        h_in, feat_W, feat_b, colscale, colshift, nullptr, hbuf, N);

    // ---- conv layers with residual
    for (int i = 0; i < L; ++i) {
        zero(sum, 2L * DH);
        bn_stats_kernel<<<256, DH, 0, stream>>>(hbuf, N, sum, sumsq);
        bn_finalize_kernel<<<1, DH, 0, stream>>>(sum, sumsq, conv_bn_g + i * DH, conv_bn_b + i * DH,
                                                 N, colscale, colshift);
        zero(msg, (long)N * DH);
        // norm_dst folded into scatter factor -> GEMM is plain + residual
        scatter_kernel<<<(E + 7) / 8, 256, 0, stream>>>(hbuf, src, dst, colscale, colshift,
                                                        norm_src, norm_dst, msg, E);
        // h += relu(msg @ conv_W[i] + conv_b[i])
        gemm_wmma_kernel<false, true><<<N / 16, 256, 0, stream>>>(
            msg, conv_W + (long)i * DH * DH, conv_b + i * DH,
            nullptr, nullptr, hbuf, hbuf, N);
    }

    // ---- graph sum pooling
    zero(hg, (long)G * DH);
    pool_kernel<<<(N + 7) / 8, 256, 0, stream>>>(hbuf, gid, hg, N);

    // ---- head: hg2 = relu(BN(hg) @ fc_W + fc_b)
    zero(sum, 2L * DH);
    bn_stats_kernel<<<8, DH, 0, stream>>>(hg, G, sum, sumsq);
    bn_finalize_kernel<<<1, DH, 0, stream>>>(sum, sumsq, fc_bn_g, fc_bn_b, G, colscale, colshift);
    gemm_wmma_kernel<true, false><<<G / 16, 256, 0, stream>>>(
        hg, fc_W, fc_b, colscale, colshift, nullptr, hg2, G);

    // ---- out = BN(hg2) @ cls_W + cls_b
    zero(sum, 2L * DH);
    bn_stats_kernel<<<8, DH, 0, stream>>>(hg2, G, sum, sumsq);
    bn_finalize_kernel<<<1, DH, 0, stream>>>(sum, sumsq, bn_hidden_g, bn_hidden_b, G, colscale, colshift);
    head_kernel<<<G, 16, 0, stream>>>(hg2, colscale, colshift, cls_W, cls_b, (float*)d_out, G);
}